// NeighbourGraphConvolution_70068096467658
// MI455X (gfx1250) — compile-verified
//
#include <hip/hip_runtime.h>
#include <math.h>

#define DDIM 128
#define BETA 0.001f
#define EPSN 1e-12f

typedef __attribute__((ext_vector_type(2))) float v2f;
typedef __attribute__((ext_vector_type(8))) float v8f;

// ---------------------------------------------------------------------------
// Kernel 1: support = input @ weight  via V_WMMA_F32_16X16X4_F32
// Block = 256 threads = 8 waves; each wave computes a 16x128 output tile.
// weight (128x128 fp32 = 64KB) staged in LDS once per block.
// ---------------------------------------------------------------------------
__global__ __launch_bounds__(256) void gcn_gemm_wmma(
    const float* __restrict__ input,
    const float* __restrict__ weight,
    float* __restrict__ support,
    int n_nodes)
{
    __shared__ float w_lds[DDIM * DDIM];  // 64 KB, k-major (same as memory)

    // cooperative coalesced load of the weight matrix into LDS
    for (int i = threadIdx.x; i < DDIM * DDIM; i += 256) w_lds[i] = weight[i];
    __syncthreads();

    const int wave = threadIdx.x >> 5;
    const int lane = threadIdx.x & 31;
    const int l15  = lane & 15;
    const int hi   = lane >> 4;      // 0: K pair {0,1}, 1: K pair {2,3}

    const int m0 = (blockIdx.x * 8 + wave) * 16;
    if (m0 >= n_nodes) return;       // wave-uniform exit (after barrier)

    v8f acc[8];
    #pragma unroll
    for (int n = 0; n < 8; ++n) acc[n] = (v8f){};

    const float* arow = input + (size_t)(m0 + l15) * DDIM;

    for (int k0 = 0; k0 < DDIM; k0 += 4) {
        const int ka = k0 + 2 * hi;                       // even -> 8B aligned
        v2f a = *reinterpret_cast<const v2f*>(arow + ka); // A[l15][ka..ka+1]
        #pragma unroll
        for (int n = 0; n < 8; ++n) {
            v2f b;
            b.x = w_lds[ka * DDIM + n * 16 + l15];        // B[ka  ][n]
            b.y = w_lds[(ka + 1) * DDIM + n * 16 + l15];  // B[ka+1][n]
            acc[n] = __builtin_amdgcn_wmma_f32_16x16x4_f32(
                false, a, false, b, (short)0, acc[n], false, false);
        }
    }

    // D layout: VGPR r holds row (r + 8*hi), column n*16 + l15
    #pragma unroll
    for (int n = 0; n < 8; ++n) {
        #pragma unroll
        for (int r = 0; r < 8; ++r) {
            support[(size_t)(m0 + r + 8 * hi) * DDIM + n * 16 + l15] = acc[n][r];
        }
    }
}

// ---------------------------------------------------------------------------
// Kernel 2: zero the aggregation buffer (graph-capture safe, deterministic)
// ---------------------------------------------------------------------------
__global__ void gcn_zero(float4* __restrict__ p, int n4)
{
    int i = blockIdx.x * blockDim.x + threadIdx.x;
    if (i < n4) p[i] = make_float4(0.f, 0.f, 0.f, 0.f);
}

// ---------------------------------------------------------------------------
// Kernel 3: per-edge gather/scale/scatter-add.
// One wave per edge: 32 lanes x float4 = the full 128-float row (coalesced
// 512B gather, support is L2-resident), then 4 f32 atomics per lane.
// ---------------------------------------------------------------------------
__global__ __launch_bounds__(256) void gcn_edge_scatter(
    const float* __restrict__ support,
    const int* __restrict__ erows,
    const int* __restrict__ ecols,
    const float* __restrict__ evals,
    float* __restrict__ agg,
    int n_edges)
{
    const int e = blockIdx.x * 8 + (threadIdx.x >> 5);
    if (e >= n_edges) return;
    const int lane = threadIdx.x & 31;

    const int   c = ecols[e];
    const int   r = erows[e];
    const float v = evals[e];

    const float4 s = *reinterpret_cast<const float4*>(
        support + (size_t)c * DDIM + lane * 4);
    float* dst = agg + (size_t)r * DDIM + lane * 4;
    atomicAdd(dst + 0, v * s.x);
    atomicAdd(dst + 1, v * s.y);
    atomicAdd(dst + 2, v * s.z);
    atomicAdd(dst + 3, v * s.w);
}

// ---------------------------------------------------------------------------
// Kernel 4: out = normalize(BETA*input + (1-BETA)*agg) + bias
// One wave per row; wave32 shuffle reduction for the row L2 norm.
// ---------------------------------------------------------------------------
__global__ __launch_bounds__(256) void gcn_finalize(
    const float* __restrict__ input,
    const float* __restrict__ agg,
    const float* __restrict__ bias,
    float* __restrict__ out,
    int n_nodes)
{
    const int row = blockIdx.x * 8 + (threadIdx.x >> 5);
    if (row >= n_nodes) return;
    const int lane = threadIdx.x & 31;
    const size_t off = (size_t)row * DDIM + lane * 4;

    const float4 in = *reinterpret_cast<const float4*>(input + off);
    const float4 ag = *reinterpret_cast<const float4*>(agg + off);

    float4 o;
    o.x = BETA * in.x + (1.0f - BETA) * ag.x;
    o.y = BETA * in.y + (1.0f - BETA) * ag.y;
    o.z = BETA * in.z + (1.0f - BETA) * ag.z;
    o.w = BETA * in.w + (1.0f - BETA) * ag.w;

    float ss = o.x * o.x + o.y * o.y + o.z * o.z + o.w * o.w;
    ss += __shfl_xor(ss, 16, 32);
    ss += __shfl_xor(ss, 8, 32);
    ss += __shfl_xor(ss, 4, 32);
    ss += __shfl_xor(ss, 2, 32);
    ss += __shfl_xor(ss, 1, 32);

    const float inv = 1.0f / fmaxf(sqrtf(ss), EPSN);
    const float4 b = *reinterpret_cast<const float4*>(bias + lane * 4);

    float4 res;
    res.x = o.x * inv + b.x;
    res.y = o.y * inv + b.y;
    res.z = o.z * inv + b.z;
    res.w = o.w * inv + b.w;
    *reinterpret_cast<float4*>(out + off) = res;
}

// ---------------------------------------------------------------------------
extern "C" void kernel_launch(void* const* d_in, const int* in_sizes, int n_in,
                              void* d_out, int out_size, void* d_ws, size_t ws_size,
                              hipStream_t stream)
{
    const float* input  = (const float*)d_in[0];
    const int*   erows  = (const int*)  d_in[1];
    const int*   ecols  = (const int*)  d_in[2];
    const float* evals  = (const float*)d_in[3];
    const float* weight = (const float*)d_in[4];
    const float* bias   = (const float*)d_in[5];
    float* out = (float*)d_out;

    const int n_nodes = in_sizes[0] / DDIM;
    const int n_edges = in_sizes[1];

    // d_out doubles as the scratch for `support` (fully overwritten by
    // gcn_finalize at the end); d_ws holds only `agg` (51.2 MB).
    float* support = out;
    float* agg = (float*)d_ws;

    // 1) zero agg
    const int n4 = (n_nodes * DDIM) / 4;
    gcn_zero<<<(n4 + 255) / 256, 256, 0, stream>>>((float4*)agg, n4);

    // 2) support = input @ weight  (WMMA)
    const int m_tiles = (n_nodes + 15) / 16;
    gcn_gemm_wmma<<<(m_tiles + 7) / 8, 256, 0, stream>>>(input, weight, support, n_nodes);

    // 3) edge gather + atomic scatter-add
    gcn_edge_scatter<<<(n_edges + 7) / 8, 256, 0, stream>>>(
        support, erows, ecols, evals, agg, n_edges);

    // 4) blend + L2 normalize + bias -> d_out (overwrites support)
    gcn_finalize<<<(n_nodes + 7) / 8, 256, 0, stream>>>(input, agg, bias, out, n_nodes);
}